// PytorchMambaSSM_6270652253004
// MI455X (gfx1250) — compile-verified
//
#include <hip/hip_runtime.h>
#include <hip/hip_bf16.h>
#include <math.h>

#define L_SEQ   1024
#define DIN     5120
#define NSTATE  16
#define RANK    160
#define EDIM    (RANK + 2 * NSTATE)   // 192
#define TCHUNK  64
#define NCHUNK  (L_SEQ / TCHUNK)      // 16

typedef __attribute__((ext_vector_type(2))) float v2f;
typedef __attribute__((ext_vector_type(8))) float v8f;

__device__ __forceinline__ v2f ld2(const float* p) { return *(const v2f*)p; }

// ---- gfx1250 async global->LDS path (guarded; sync fallback keeps compile safe)
#if __has_builtin(__builtin_amdgcn_global_load_async_to_lds_b128)
#define USE_ASYNC_LDS 1
// Builtin expects: (AS1 ptr to int __vector(4), AS3 ptr to int __vector(4), imm, imm)
typedef int v4i_g __attribute__((vector_size(16)));
typedef __attribute__((address_space(1))) v4i_g glob_v4i;
typedef __attribute__((address_space(3))) v4i_g lds_v4i;
#endif

__device__ __forceinline__ void async_copy16(const float* gsrc, float* ldst) {
#ifdef USE_ASYNC_LDS
    __builtin_amdgcn_global_load_async_to_lds_b128(
        (glob_v4i*)gsrc, (lds_v4i*)ldst, 0, 0);
#else
    *(float4*)ldst = *(const float4*)gsrc;
#endif
}

__device__ __forceinline__ void wait_async_lds() {
#ifdef USE_ASYNC_LDS
#if __has_builtin(__builtin_amdgcn_s_wait_asynccnt)
    __builtin_amdgcn_s_wait_asynccnt(0);
#else
    asm volatile("s_wait_asynccnt 0x0" ::: "memory");
#endif
#endif
}

// ---------------------------------------------------------------------------
// Kernel 0: A2[d,n] = -exp(A_log[d,n]) * log2(e)   (fold log2e so the scan's
// exp(delta*A) becomes a single v_exp_f32 = exp2)
// ---------------------------------------------------------------------------
__global__ void prep_A2(const float* __restrict__ A_log, float* __restrict__ A2, int n) {
    int i = blockIdx.x * blockDim.x + threadIdx.x;
    if (i < n) A2[i] = -expf(A_log[i]) * 1.44269504088896340736f;
}

// ---------------------------------------------------------------------------
// Kernel 1: x_dbl[l,e] = sum_d x[l,d] * w[e,d]    (M=1024, N=192, K=5120)
// f32 WMMA 16x16x4. Block = 3 waves; wave w covers N-tiles 4w..4w+3.
// ---------------------------------------------------------------------------
__global__ void __launch_bounds__(96)
gemm_xproj(const float* __restrict__ x, const float* __restrict__ w,
           float* __restrict__ xdbl) {
    const int lane  = threadIdx.x & 31;
    const int wave  = threadIdx.x >> 5;
    const int half  = lane >> 4;
    const int lm    = lane & 15;
    const int tileM = blockIdx.x * 16;
    const int row   = tileM + lm;

    v8f acc[4];
#pragma unroll
    for (int j = 0; j < 4; ++j)
#pragma unroll
        for (int r = 0; r < 8; ++r) acc[j][r] = 0.0f;

    const float* aRow  = x + (size_t)row * DIN + 2 * half;
    const float* bBase = w + 2 * half;

    for (int kk = 0; kk < DIN; kk += 4) {
        v2f a = ld2(aRow + kk);
#pragma unroll
        for (int j = 0; j < 4; ++j) {
            const int col = (wave * 4 + j) * 16 + lm;
            v2f b = ld2(bBase + (size_t)col * DIN + kk);
            acc[j] = __builtin_amdgcn_wmma_f32_16x16x4_f32(
                false, a, false, b, (short)0, acc[j], false, false);
        }
    }

#pragma unroll
    for (int j = 0; j < 4; ++j) {
        const int col = (wave * 4 + j) * 16 + lm;
#pragma unroll
        for (int r = 0; r < 8; ++r) {
            const int orow = tileM + r + 8 * half;
            xdbl[(size_t)orow * EDIM + col] = acc[j][r];
        }
    }
}

// ---------------------------------------------------------------------------
// Kernel 2: delta[l,d] = softplus( sum_r xdbl[l,r] * dtw[d,r] + dtb[d] )
// M=1024, N=5120, K=160.  Block = 4 waves x 4 N-tiles = 256 columns.
// ---------------------------------------------------------------------------
__global__ void __launch_bounds__(128)
gemm_dt(const float* __restrict__ xdbl, const float* __restrict__ dtw,
        const float* __restrict__ dtb, float* __restrict__ delta) {
    const int lane   = threadIdx.x & 31;
    const int wave   = threadIdx.x >> 5;
    const int half   = lane >> 4;
    const int lm     = lane & 15;
    const int tileM  = blockIdx.x * 16;
    const int tileNb = blockIdx.y * 256 + wave * 64;
    const int row    = tileM + lm;

    v8f acc[4];
#pragma unroll
    for (int j = 0; j < 4; ++j)
#pragma unroll
        for (int r = 0; r < 8; ++r) acc[j][r] = 0.0f;

    const float* aRow  = xdbl + (size_t)row * EDIM + 2 * half;  // dt = cols 0..159
    const float* bBase = dtw + 2 * half;

    for (int kk = 0; kk < RANK; kk += 4) {
        v2f a = ld2(aRow + kk);
#pragma unroll
        for (int j = 0; j < 4; ++j) {
            const int col = tileNb + j * 16 + lm;
            v2f b = ld2(bBase + (size_t)col * RANK + kk);
            acc[j] = __builtin_amdgcn_wmma_f32_16x16x4_f32(
                false, a, false, b, (short)0, acc[j], false, false);
        }
    }

#pragma unroll
    for (int j = 0; j < 4; ++j) {
        const int col  = tileNb + j * 16 + lm;
        const float bv = dtb[col];
#pragma unroll
        for (int r = 0; r < 8; ++r) {
            const int orow = tileM + r + 8 * half;
            float v = acc[j][r] + bv;
            float sp = (v > 20.0f) ? v : log1pf(expf(v));   // softplus
            delta[(size_t)orow * DIN + col] = sp;
        }
    }
}

// ---------------------------------------------------------------------------
// Kernel 3: fused selective scan, double-buffered async B/C staging.
// Thread d holds h[0..15] in VGPRs; one barrier + one asynccnt wait per chunk.
// ---------------------------------------------------------------------------
__device__ __forceinline__ void stage_chunk(const float* __restrict__ xdbl,
                                            int c, float* dstBuf, int tid) {
    // 64 rows x 32 floats = 512 x 16B transfers; 256 threads issue 2 each
#pragma unroll
    for (int q = 0; q < 2; ++q) {
        const int lin  = q * 256 + tid;   // 0..511
        const int tt   = lin >> 3;        // row in chunk
        const int part = lin & 7;         // 16B piece within the 128B row
        const float* src = xdbl + (size_t)(c * TCHUNK + tt) * EDIM + RANK + part * 4;
        async_copy16(src, dstBuf + tt * 32 + part * 4);
    }
}

__global__ void __launch_bounds__(256)
mamba_scan(const float* __restrict__ x, const float* __restrict__ xdbl,
           const float* __restrict__ delta, const float* __restrict__ A2,
           const float* __restrict__ Dw, float* __restrict__ out) {
    __shared__ __align__(16) float sBC[2][TCHUNK * 32];   // 2 x 8KB

    const int tid = threadIdx.x;
    const int d   = blockIdx.x * 256 + tid;

    float a2[NSTATE];
#pragma unroll
    for (int q = 0; q < 4; ++q)
        ((float4*)a2)[q] = ((const float4*)(A2 + (size_t)d * NSTATE))[q];

    const float Dv = Dw[d];

    float h[NSTATE];
#pragma unroll
    for (int n = 0; n < NSTATE; ++n) h[n] = 0.0f;

    stage_chunk(xdbl, 0, sBC[0], tid);   // prologue: chunk 0 in flight

    for (int c = 0; c < NCHUNK; ++c) {
        const int cur = c & 1;
        wait_async_lds();        // my waves' async writes for chunk c done
        __syncthreads();         // all waves' writes visible; prev reads done

        if (c + 1 < NCHUNK)      // overlap: next chunk DMA during compute
            stage_chunk(xdbl, c + 1, sBC[cur ^ 1], tid);

#pragma unroll 1
        for (int tl = 0; tl < TCHUNK; ++tl) {
            const int t = c * TCHUNK + tl;
            const float dl = delta[(size_t)t * DIN + d];
            const float xv = x[(size_t)t * DIN + d];

            if (t + TCHUNK < L_SEQ) {   // gfx1250 global_prefetch_b8 lookahead
                __builtin_prefetch(delta + (size_t)(t + TCHUNK) * DIN + d, 0, 0);
                __builtin_prefetch(x + (size_t)(t + TCHUNK) * DIN + d, 0, 0);
            }

            float bc[32];
#pragma unroll
            for (int q = 0; q < 8; ++q)
                ((float4*)bc)[q] = ((const float4*)(sBC[cur] + tl * 32))[q];

            const float dx = dl * xv;
            float y = 0.0f;
#pragma unroll
            for (int n = 0; n < NSTATE; ++n) {
                const float dA = __builtin_exp2f(dl * a2[n]);       // exp(delta*A)
                h[n] = __builtin_fmaf(dA, h[n], dx * bc[n]);        // h = dA*h + dBu
                y    = __builtin_fmaf(h[n], bc[16 + n], y);         // y += h*C
            }
            out[(size_t)t * DIN + d] = y + xv * Dv;
        }
    }
}

// ---------------------------------------------------------------------------
extern "C" void kernel_launch(void* const* d_in, const int* in_sizes, int n_in,
                              void* d_out, int out_size, void* d_ws, size_t ws_size,
                              hipStream_t stream) {
    const float* x    = (const float*)d_in[0];
    const float* Alog = (const float*)d_in[1];
    const float* Dw   = (const float*)d_in[2];
    const float* xpw  = (const float*)d_in[3];
    const float* dtw  = (const float*)d_in[4];
    const float* dtb  = (const float*)d_in[5];
    float* out = (float*)d_out;

    float* ws    = (float*)d_ws;
    float* xdbl  = ws;                               // 1024*192 floats
    float* delta = xdbl + (size_t)L_SEQ * EDIM;      // 1024*5120 floats
    float* A2    = delta + (size_t)L_SEQ * DIN;      // 5120*16 floats

    prep_A2<<<(DIN * NSTATE + 255) / 256, 256, 0, stream>>>(Alog, A2, DIN * NSTATE);
    gemm_xproj<<<L_SEQ / 16, 96, 0, stream>>>(x, xpw, xdbl);
    gemm_dt<<<dim3(L_SEQ / 16, DIN / 256), 128, 0, stream>>>(xdbl, dtw, dtb, delta);
    mamba_scan<<<DIN / 256, 256, 0, stream>>>(x, xdbl, delta, A2, Dw, out);
}